// ProbSparseMultiheadAttention_75170517615178
// MI455X (gfx1250) — compile-verified
//
#include <hip/hip_runtime.h>
#include <math.h>
#include <stdint.h>

typedef float v2f __attribute__((ext_vector_type(2)));
typedef float v4f __attribute__((ext_vector_type(4)));
typedef float v8f __attribute__((ext_vector_type(8)));

#define B_    2
#define S_    2048
#define H_    16
#define DK_   64
#define DM_   1024
#define TOPK_ 16

// ------------------------------------------------------------------
// Async global->LDS copy (CDNA5 GLOBAL_LOAD_ASYNC_TO_LDS_B128, ASYNCcnt)
// Builtin signature (from clang diagnostic): (v4i AS1*, v4i AS3*, Ii, Ii)
// ------------------------------------------------------------------
#if __has_builtin(__builtin_amdgcn_global_load_async_to_lds_b128)
#define HAVE_ASYNC 1
#else
#define HAVE_ASYNC 0
#endif

typedef int v4i_ __attribute__((vector_size(4 * sizeof(int))));
typedef __attribute__((address_space(1))) v4i_* g4p;
typedef __attribute__((address_space(3))) v4i_* l4p;

__device__ __forceinline__ void cp16_g2l(const float* g, float* l)
{
#if HAVE_ASYNC
  __builtin_amdgcn_global_load_async_to_lds_b128(
      (g4p)(uintptr_t)g,
      (l4p)(unsigned)(uintptr_t)l,   // AS3 pointers are 32-bit; low bits = LDS offset
      0, 0);
#else
  *(float4*)l = *(const float4*)g;
#endif
}

#if HAVE_ASYNC && __has_builtin(__builtin_amdgcn_s_wait_asynccnt)
#define ASYNC_WAIT(n) __builtin_amdgcn_s_wait_asynccnt(n)
#elif HAVE_ASYNC
#define ASYNC_WAIT(n) asm volatile("s_wait_asynccnt %0" :: "n"(n) : "memory")
#else
#define ASYNC_WAIT(n) ((void)0)
#endif

// =====================================================================
// GEMM: C[M,N] = A[M,K] * W[N,K]^T + bias[N]
// mode 0: C row-major [M,N]
// mode 1: head-split output C[((b*H + h)*S + s)*DK + d], row=b*S+s, col=h*DK+d
// Block tile 32x64, 8 waves, 16x16 fp32 WMMA accumulators,
// double-buffered LDS fed by async global->LDS copies.
// =====================================================================
#define TM  32
#define TN  64
#define KB  32
#define KBP (KB + 4)

__global__ __launch_bounds__(256) void gemm_xwT(
    const float* __restrict__ A, const float* __restrict__ W,
    const float* __restrict__ bias, float* __restrict__ C,
    int M, int N, int K, int mode)
{
  __shared__ float As[2][TM][KBP];
  __shared__ float Ws[2][TN][KBP];

  const int bm   = blockIdx.x * TM;
  const int bn   = blockIdx.y * TN;
  const int tid  = threadIdx.x;
  const int wave = tid >> 5;
  const int lane = tid & 31;
  const int wm   = (wave >> 2) << 4;   // 0 or 16
  const int wn   = (wave & 3) << 4;    // 0,16,32,48
  const int hi   = lane >> 4;          // K-pair select (ISA fp32 A/B layout)
  const int lo   = lane & 15;          // row (A) / col (B)

  // per-thread copy coordinates (3 x b128 per thread per tile)
  const int ra  = tid >> 3, ca = (tid & 7) << 2;   // A: 256 float4
  const int rw0 = tid >> 3;                        // W: 512 float4, two rows
  const int rw1 = (tid + 256) >> 3;

  auto issue = [&](int buf, int k0) {
    cp16_g2l(A + (size_t)(bm + ra)  * K + (k0 + ca), &As[buf][ra][ca]);
    cp16_g2l(W + (size_t)(bn + rw0) * K + (k0 + ca), &Ws[buf][rw0][ca]);
    cp16_g2l(W + (size_t)(bn + rw1) * K + (k0 + ca), &Ws[buf][rw1][ca]);
  };

  v8f acc = {};
  const int nk = K / KB;

  issue(0, 0);                       // prime the pipeline

  for (int it = 0; it < nk; ++it) {
    const int cur = it & 1;
    if (it + 1 < nk) {
      issue(cur ^ 1, (it + 1) * KB); // overlap next tile's DMA with compute
      ASYNC_WAIT(3);                 // current tile's 3 copies complete
    } else {
      ASYNC_WAIT(0);
    }
    __syncthreads();

    #pragma unroll
    for (int kk = 0; kk < KB; kk += 4) {
      v2f a, b;
      a.x = As[cur][wm + lo][kk + 2 * hi];
      a.y = As[cur][wm + lo][kk + 2 * hi + 1];
      b.x = Ws[cur][wn + lo][kk + 2 * hi];
      b.y = Ws[cur][wn + lo][kk + 2 * hi + 1];
      acc = __builtin_amdgcn_wmma_f32_16x16x4_f32(
          false, a, false, b, (short)0, acc, false, false);
    }
    __syncthreads();                 // all waves done reading `cur` before reuse
  }

  // C layout: VGPR r -> row (r | hi<<3), col = lo
  const int col = bn + wn + lo;
  const float bv = bias[col];
  #pragma unroll
  for (int r = 0; r < 8; ++r) {
    const int row = bm + wm + (hi << 3) + r;
    const float val = acc[r] + bv;
    if (mode == 0) {
      C[(size_t)row * N + col] = val;
    } else {
      const int b = row / S_, s = row % S_;
      const int h = col / DK_, d = col % DK_;
      C[(((size_t)b * H_ + h) * S_ + s) * DK_ + d] = val;
    }
  }
}

// =====================================================================
// Attention: one block per (b, h, 16-query tile).
// scores = Q·K^T / 8 via fp32 WMMA, streaming top-16/row, softmax over
// kept values (others are exactly 0 after exp underflow), NT dense-zero +
// scatter weights to d_out, then 16-term weighted-V accumulation.
// =====================================================================
#define STRIP  128
#define NSTRIP (S_ / STRIP)   // 16
#define SEGK   8              // keys per thread per strip (128 keys / 16 segs)

__global__ __launch_bounds__(256) void attn_topk(
    const float* __restrict__ Qw, const float* __restrict__ Kw,
    const float* __restrict__ Vw, float* __restrict__ wout,
    float* __restrict__ attn)
{
  __shared__ float Qs[16][DK_ + 4];
  __shared__ float Ss[16][STRIP + 4];
  __shared__ float cand_v[16][16][TOPK_];
  __shared__ int   cand_i[16][16][TOPK_];
  __shared__ float topv_s[16][TOPK_];
  __shared__ int   topi_s[16][TOPK_];

  const int tid  = threadIdx.x;
  const int wave = tid >> 5;
  const int lane = tid & 31;
  const int hi   = lane >> 4;
  const int lo   = lane & 15;
  const int qt   = blockIdx.x % (S_ / 16);
  const int bh   = blockIdx.x / (S_ / 16);

  const float* Qbase = Qw + ((size_t)bh * S_ + (size_t)qt * 16) * DK_;
  const float* Kbase = Kw + (size_t)bh * S_ * DK_;
  const float* Vbase = Vw + (size_t)bh * S_ * DK_;

  // Stage 16x64 Q tile (256 threads x 1 float4)
  {
    const int r = tid >> 4, c4 = (tid & 15) << 2;
    const float4 v = *(const float4*)(Qbase + (size_t)r * DK_ + c4);
    *(float4*)&Qs[r][c4] = v;
  }
  __syncthreads();

  const int srow = tid >> 4;   // row this thread scans
  const int sseg = tid & 15;   // 8-key segment within strip

  float tv[TOPK_]; int ti[TOPK_];
  #pragma unroll
  for (int e = 0; e < TOPK_; ++e) { tv[e] = -INFINITY; ti[e] = 0; }

  for (int strip = 0; strip < NSTRIP; ++strip) {
    const int kb   = strip * STRIP;
    const int keyb = kb + wave * 16;

    // prefetch next strip's K rows (gfx1250 global_prefetch_b8)
    if (strip + 1 < NSTRIP)
      __builtin_prefetch(Kbase + (size_t)(keyb + STRIP + lo) * DK_, 0, 3);

    // 16x16 score tile: 16 chained fp32 WMMAs over Dk=64
    const float2* kp = (const float2*)(Kbase + (size_t)(keyb + lo) * DK_);
    v8f sc = {};
    #pragma unroll
    for (int kk = 0; kk < DK_; kk += 4) {
      v2f a, b;
      a.x = Qs[lo][kk + 2 * hi];
      a.y = Qs[lo][kk + 2 * hi + 1];
      const float2 bb = kp[(kk >> 1) + hi];
      b.x = bb.x; b.y = bb.y;
      sc = __builtin_amdgcn_wmma_f32_16x16x4_f32(
          false, a, false, b, (short)0, sc, false, false);
    }
    #pragma unroll
    for (int r = 0; r < 8; ++r)
      Ss[(hi << 3) + r][wave * 16 + lo] = sc[r] * 0.125f;
    __syncthreads();

    // streaming top-16 update over this thread's 8-key segment
    for (int j = 0; j < SEGK; ++j) {
      const float v = Ss[srow][sseg * SEGK + j];
      if (v > tv[0]) {
        int i = 0;
        while (i < TOPK_ - 1 && v > tv[i + 1]) {
          tv[i] = tv[i + 1]; ti[i] = ti[i + 1]; ++i;
        }
        tv[i] = v; ti[i] = kb + sseg * SEGK + j;
      }
    }
    __syncthreads();
  }

  // dump per-segment candidate lists
  #pragma unroll
  for (int e = 0; e < TOPK_; ++e) {
    cand_v[srow][sseg][e] = tv[e];
    cand_i[srow][sseg][e] = ti[e];
  }
  __syncthreads();

  // dense zero-fill of this block's 16 weight rows (16 x 2048 fp32),
  // nontemporal: 512MB stream must not displace L2-resident K/V.
  {
    const v4f z = {};
    float* wbase = wout + ((size_t)bh * S_ + (size_t)qt * 16) * S_;
    for (int t = tid; t < (16 * S_) / 4; t += 256)
      __builtin_nontemporal_store(z, (v4f*)(wbase + 4 * (size_t)t));
  }

  // per-row merge of 16 candidate lists + softmax over kept 16
  if (tid < 16) {
    #pragma unroll
    for (int e = 0; e < TOPK_; ++e) { tv[e] = -INFINITY; ti[e] = 0; }
    for (int seg = 0; seg < 16; ++seg) {
      for (int e = 0; e < TOPK_; ++e) {
        const float v = cand_v[tid][seg][e];
        if (v > tv[0]) {
          const int idx = cand_i[tid][seg][e];
          int i = 0;
          while (i < TOPK_ - 1 && v > tv[i + 1]) {
            tv[i] = tv[i + 1]; ti[i] = ti[i + 1]; ++i;
          }
          tv[i] = v; ti[i] = idx;
        }
      }
    }
    const float mx = tv[TOPK_ - 1];
    float w[TOPK_];
    float sum = 0.f;
    #pragma unroll
    for (int e = 0; e < TOPK_; ++e) { w[e] = __expf(tv[e] - mx); sum += w[e]; }
    const float inv = 1.f / sum;
    #pragma unroll
    for (int e = 0; e < TOPK_; ++e) {
      topv_s[tid][e] = w[e] * inv;
      topi_s[tid][e] = ti[e];
    }
  }
  __syncthreads();

  // scatter normalized top-k weights (zeros are already down)
  if (tid < 16) {
    float* wrow = wout + ((size_t)bh * S_ + (size_t)qt * 16 + tid) * S_;
    #pragma unroll
    for (int e = 0; e < TOPK_; ++e)
      wrow[topi_s[tid][e]] = topv_s[tid][e];
  }

  // attn[q,:] = sum_j w_j * V[idx_j,:]  (16 rows x 64 cols, 4 cols/thread)
  {
    const int row = tid >> 4;
    const int c4  = (tid & 15) << 2;
    float4 a4 = make_float4(0.f, 0.f, 0.f, 0.f);
    #pragma unroll
    for (int e = 0; e < TOPK_; ++e) {
      const float wgt = topv_s[row][e];
      const float4 vv =
          *(const float4*)(Vbase + (size_t)topi_s[row][e] * DK_ + c4);
      a4.x += wgt * vv.x; a4.y += wgt * vv.y;
      a4.z += wgt * vv.z; a4.w += wgt * vv.w;
    }
    const int b = bh / H_, h = bh % H_;
    float* arow = attn + ((size_t)b * S_ + (size_t)qt * 16 + row) * DM_
                + (size_t)h * DK_ + c4;
    *(float4*)arow = a4;
  }
}

// =====================================================================
extern "C" void kernel_launch(void* const* d_in, const int* in_sizes, int n_in,
                              void* d_out, int out_size, void* d_ws, size_t ws_size,
                              hipStream_t stream)
{
  (void)in_sizes; (void)n_in; (void)out_size; (void)ws_size;

  const float* query = (const float*)d_in[0];
  const float* key_  = (const float*)d_in[1];
  const float* value = (const float*)d_in[2];
  const float* Wq = (const float*)d_in[3];
  const float* bq = (const float*)d_in[4];
  const float* Wk = (const float*)d_in[5];
  const float* bk = (const float*)d_in[6];
  const float* Wv = (const float*)d_in[7];
  const float* bv = (const float*)d_in[8];
  const float* Wo = (const float*)d_in[9];
  const float* bo = (const float*)d_in[10];

  float* out = (float*)d_out;
  const size_t QKV = (size_t)B_ * H_ * S_ * DK_;       // 4M floats each
  float* Qw   = (float*)d_ws;
  float* Kw   = Qw + QKV;
  float* Vw   = Kw + QKV;
  float* attn = Vw + QKV;                              // [B,S,DM] row-major
  float* weights = out + (size_t)B_ * S_ * DM_;        // second tuple output

  const int M = B_ * S_;                               // 4096
  dim3 gg(M / TM, DM_ / TN);                           // 128 x 16

  // Q/K/V projections into head-split layout [B,H,S,Dk]
  gemm_xwT<<<gg, 256, 0, stream>>>(query, Wq, bq, Qw, M, DM_, DM_, 1);
  gemm_xwT<<<gg, 256, 0, stream>>>(key_,  Wk, bk, Kw, M, DM_, DM_, 1);
  gemm_xwT<<<gg, 256, 0, stream>>>(value, Wv, bv, Vw, M, DM_, DM_, 1);

  // scores + top-k + softmax + weights output + sparse weights·V
  attn_topk<<<B_ * H_ * (S_ / 16), 256, 0, stream>>>(Qw, Kw, Vw, weights, attn);

  // out = attn · Wo^T + bo
  gemm_xwT<<<gg, 256, 0, stream>>>(attn, Wo, bo, out, M, DM_, DM_, 0);
}